// ConditionalOT_51264729645654
// MI455X (gfx1250) — compile-verified
//
#include <hip/hip_runtime.h>
#include <math.h>

typedef float v2f __attribute__((ext_vector_type(2)));
typedef float v8f __attribute__((ext_vector_type(8)));

#define BN 4096
#define DN 256
#define EPSV 0.1f
#define INV_EPS 10.0f
#define N_ITER 50
#define NRB 32                    // row chunks for column partials
#define ROWS_PER_CHUNK (BN / NRB) // 128

// ---------- online logsumexp helpers ----------
__device__ __forceinline__ void lse_online(float x, float& m, float& s) {
    if (x > m) { s = s * __expf(m - x) + 1.0f; m = x; }
    else       { s += __expf(x - m); }
}
__device__ __forceinline__ void lse_combine(float m2, float s2, float& m, float& s) {
    float M = fmaxf(m, m2);
    s = s * __expf(m - M) + s2 * __expf(m2 - M);
    m = M;
}

// ---------- squared row norms of zc / zp ----------
__global__ void cot_sqnorm(const float* __restrict__ zc, const float* __restrict__ zp,
                           float* __restrict__ sqc, float* __restrict__ sqp) {
    __shared__ float sh[256];
    const int row = blockIdx.x;
    const float* src = (blockIdx.y == 0) ? zc : zp;
    float val = src[row * DN + threadIdx.x];
    sh[threadIdx.x] = val * val;
    __syncthreads();
    for (int off = 128; off > 0; off >>= 1) {
        if (threadIdx.x < off) sh[threadIdx.x] += sh[threadIdx.x + off];
        __syncthreads();
    }
    if (threadIdx.x == 0) ((blockIdx.y == 0) ? sqc : sqp)[row] = sh[0];
}

// ---------- zero u, v ----------
__global__ void cot_init_uv(float* __restrict__ u, float* __restrict__ v) {
    int t = blockIdx.x * 256 + threadIdx.x;   // 8192 threads
    if (t < BN) u[t] = 0.0f;
    else        v[t - BN] = 0.0f;
}

// ---------- cost matrix: Cmat = max(sqc_i + sqp_j - 2*zc.zp^T, 0) via f32 WMMA ----------
// 8 waves per block, one 16x16 tile per wave; tile grid 256x256.
__global__ void cot_cost_gemm(const float* __restrict__ zc, const float* __restrict__ zp,
                              const float* __restrict__ sqc, const float* __restrict__ sqp,
                              float* __restrict__ Cmat) {
    const int wave = threadIdx.x >> 5;
    const int lane = threadIdx.x & 31;
    const int tile = blockIdx.x * 8 + wave;     // 8192 blocks -> 65536 tiles
    const int ti = tile >> 8;                   // 0..255
    const int tj = tile & 255;
    const int m  = lane & 15;
    const int kh = (lane >> 4) << 1;            // 0 or 2
    const float* arow = zc + (ti * 16 + m) * DN + kh;
    const float* brow = zp + (tj * 16 + m) * DN + kh;   // B = zp^T
    v8f acc = {0.f,0.f,0.f,0.f,0.f,0.f,0.f,0.f};
    for (int k = 0; k < DN; k += 4) {
        v2f a = *(const v2f*)(arow + k);
        v2f b = *(const v2f*)(brow + k);
        acc = __builtin_amdgcn_wmma_f32_16x16x4_f32(false, a, false, b,
                                                    (short)0, acc, false, false);
    }
    // C/D layout: VGPR r -> row (r + 8*(lane>=16)), col = lane&15
    const int col = tj * 16 + (lane & 15);
    const int row_base = ti * 16 + ((lane >> 4) << 3);
    const float sp = sqp[col];
    for (int r = 0; r < 8; ++r) {
        const int row = row_base + r;
        float c = sqc[row] + sp - 2.0f * acc[r];
        Cmat[(size_t)row * BN + col] = fmaxf(c, 0.0f);
    }
}

// ---------- u-update: one block per row ----------
__global__ void cot_row_update(const float* __restrict__ Cmat, float* __restrict__ u,
                               const float* __restrict__ v, float log_a) {
    __shared__ float sm[256], ss[256];
    const int i = blockIdx.x;
    const int tid = threadIdx.x;
    const float ui = u[i];
    const float* crow = Cmat + (size_t)i * BN;
    float m = -INFINITY, s = 0.0f;
    for (int j = tid; j < BN; j += 256) {
        float x = (ui + v[j] - crow[j]) * INV_EPS;
        lse_online(x, m, s);
    }
    sm[tid] = m; ss[tid] = s;
    __syncthreads();
    for (int off = 128; off > 0; off >>= 1) {
        if (tid < off) {
            float mm = sm[tid], sv = ss[tid];
            lse_combine(sm[tid + off], ss[tid + off], mm, sv);
            sm[tid] = mm; ss[tid] = sv;
        }
        __syncthreads();
    }
    if (tid == 0) u[i] = EPSV * (log_a - (sm[0] + __logf(ss[0])));
}

// ---------- v-update phase 1: coalesced column partials ----------
// grid (16, NRB): blockIdx.x -> 256-column stripe, blockIdx.y -> 128-row chunk
__global__ void cot_col_partial(const float* __restrict__ Cmat, const float* __restrict__ u,
                                const float* __restrict__ v,
                                float* __restrict__ pm, float* __restrict__ ps) {
    __shared__ float ush[ROWS_PER_CHUNK];
    const int tid = threadIdx.x;
    const int r0 = blockIdx.y * ROWS_PER_CHUNK;
    if (tid < ROWS_PER_CHUNK) ush[tid] = u[r0 + tid];
    __syncthreads();
    const int j = blockIdx.x * 256 + tid;
    const float vj = v[j];
    const float* cp = Cmat + (size_t)r0 * BN + j;
    float m = -INFINITY, s = 0.0f;
    for (int r = 0; r < ROWS_PER_CHUNK; ++r) {
        float x = (ush[r] + vj - cp[(size_t)r * BN]) * INV_EPS;
        lse_online(x, m, s);
    }
    pm[blockIdx.y * BN + j] = m;
    ps[blockIdx.y * BN + j] = s;
}

// ---------- v-update phase 2: combine partials ----------
__global__ void cot_col_final(const float* __restrict__ pm, const float* __restrict__ ps,
                              float* __restrict__ v, float log_b) {
    const int j = blockIdx.x * 256 + threadIdx.x;
    float m = -INFINITY, s = 0.0f;
    for (int r = 0; r < NRB; ++r)
        lse_combine(pm[r * BN + j], ps[r * BN + j], m, s);
    v[j] = EPSV * (log_b - (m + __logf(s)));
}

// ---------- P = exp((u + v - C)/eps), in place over the C region ----------
__global__ void cot_p_final(float* __restrict__ Pmat, const float* __restrict__ u,
                            const float* __restrict__ v) {
    const int t = blockIdx.x * 256 + threadIdx.x;
    const int idx = t * 4;                 // 4 consecutive elems, never crosses a row
    const int i = idx >> 12;
    const int j = idx & (BN - 1);
    const float ui = u[i];
    float4 c = *(const float4*)(Pmat + idx);
    float4 r;
    r.x = __expf((ui + v[j + 0] - c.x) * INV_EPS);
    r.y = __expf((ui + v[j + 1] - c.y) * INV_EPS);
    r.z = __expf((ui + v[j + 2] - c.z) * INV_EPS);
    r.w = __expf((ui + v[j + 3] - c.w) * INV_EPS);
    *(float4*)(Pmat + idx) = r;
}

// ---------- z_pred = P @ zp via f32 WMMA: M=4096, K=4096, N=256 ----------
__global__ void cot_pred_gemm(const float* __restrict__ Pmat, const float* __restrict__ zp,
                              float* __restrict__ out) {
    const int wave = threadIdx.x >> 5;
    const int lane = threadIdx.x & 31;
    const int tile = blockIdx.x * 8 + wave;   // 512 blocks -> 4096 tiles
    const int ti = tile >> 4;                 // 0..255 row tile
    const int tj = tile & 15;                 // 0..15  col tile
    const int m  = lane & 15;
    const int kh = (lane >> 4) << 1;
    const float* arow = Pmat + (size_t)(ti * 16 + m) * BN + kh;
    const int col = tj * 16 + m;
    v8f acc = {0.f,0.f,0.f,0.f,0.f,0.f,0.f,0.f};
    for (int k = 0; k < BN; k += 4) {
        v2f a = *(const v2f*)(arow + k);
        v2f b;
        b.x = zp[(size_t)(k + kh + 0) * DN + col];
        b.y = zp[(size_t)(k + kh + 1) * DN + col];
        acc = __builtin_amdgcn_wmma_f32_16x16x4_f32(false, a, false, b,
                                                    (short)0, acc, false, false);
    }
    const int ocol = tj * 16 + (lane & 15);
    const int row_base = ti * 16 + ((lane >> 4) << 3);
    for (int r = 0; r < 8; ++r)
        out[(size_t)(row_base + r) * DN + ocol] = acc[r];
}

extern "C" void kernel_launch(void* const* d_in, const int* in_sizes, int n_in,
                              void* d_out, int out_size, void* d_ws, size_t ws_size,
                              hipStream_t stream) {
    const float* zc = (const float*)d_in[0];   // [4096, 256]
    const float* zp = (const float*)d_in[1];   // [4096, 256]
    // d_in[2] ("p") unused by the reference.

    float* out   = (float*)d_out;
    float* zpred = out;                 // [4096, 256]
    float* Pmat  = out + BN * DN;       // [4096, 4096]; holds C, then P in place

    float* u   = (float*)d_ws;          // [4096]
    float* v   = u + BN;                // [4096]
    float* sqc = v + BN;                // [4096]
    float* sqp = sqc + BN;              // [4096]
    float* pm  = sqp + BN;              // [NRB, 4096]
    float* ps  = pm + NRB * BN;         // [NRB, 4096]

    const float log_ab = logf(1.0f / (float)BN + 1e-8f);

    cot_sqnorm<<<dim3(BN, 2), 256, 0, stream>>>(zc, zp, sqc, sqp);
    cot_cost_gemm<<<8192, 256, 0, stream>>>(zc, zp, sqc, sqp, Pmat);
    cot_init_uv<<<32, 256, 0, stream>>>(u, v);

    for (int it = 0; it < N_ITER; ++it) {
        cot_row_update<<<BN, 256, 0, stream>>>(Pmat, u, v, log_ab);
        cot_col_partial<<<dim3(16, NRB), 256, 0, stream>>>(Pmat, u, v, pm, ps);
        cot_col_final<<<16, 256, 0, stream>>>(pm, ps, v, log_ab);
    }

    cot_p_final<<<(BN * BN) / (4 * 256), 256, 0, stream>>>(Pmat, u, v);
    cot_pred_gemm<<<512, 256, 0, stream>>>(Pmat, zp, zpred);
}